// KronyMLP_25443386261855
// MI455X (gfx1250) — compile-verified
//
#include <hip/hip_runtime.h>
#include <math.h>

typedef __attribute__((ext_vector_type(2))) float v2f;
typedef __attribute__((ext_vector_type(4))) float v4f;
typedef __attribute__((ext_vector_type(8))) float v8f;

#define DM    1536   // d_model
#define N1    32     // c_fc_1 cols / c_proj_1 rows
#define NJ    12     // kron factor length
#define KB    128    // k-block for GEMM1
#define NBLK  128    // n-block for GEMM2
#define WAVES 8      // waves per workgroup (wave32)
#define XSTR  132    // x LDS row stride (dwords, padded, 16B-aligned)
#define WTSTR 132    // W1^T LDS row stride: [32 n][KB k] padded
#define GSTR  36     // g LDS row stride (dwords, padded)
#define PTSTR 36     // P^T LDS row stride: [NBLK n][32 k] padded

__device__ __forceinline__ float gelu_erf(float z) {
    // exact GELU: 0.5*z*(1+erf(z/sqrt(2)))
    return 0.5f * z * (1.0f + erff(z * 0.70710678118654752f));
}

__global__ __launch_bounds__(256, 1)
void krony_mlp_kernel(const float* __restrict__ x,
                      const float* __restrict__ w1,   // [1536][32]
                      const float* __restrict__ f2,   // [12]
                      const float* __restrict__ p1,   // [32][1536]
                      const float* __restrict__ p2,   // [12]
                      float* __restrict__ out)
{
    extern __shared__ float smem[];
    float* lds_x  = smem;                          // [WAVES][16][XSTR]
    float* lds_wt = lds_x + WAVES * 16 * XSTR;     // [32][WTSTR]  (W1 chunk, transposed)
    float* lds_g  = lds_wt + N1 * WTSTR;           // [WAVES][16][GSTR]
    float* lds_pt = lds_g + WAVES * 16 * GSTR;     // [NBLK][PTSTR] (P chunk, transposed)

    const int tid  = threadIdx.x;
    const int wave = tid >> 5;
    const int lane = tid & 31;
    const int half = lane >> 4;     // 0: lanes 0-15, 1: lanes 16-31
    const int l16  = lane & 15;
    const int tok0 = (blockIdx.x * WAVES + wave) * 16;

    float* xw = lds_x + wave * 16 * XSTR;
    float* gw = lds_g + wave * 16 * GSTR;

    // small kron vectors into registers (uniform -> scalar loads)
    float f2r[NJ], p2r[NJ];
#pragma unroll
    for (int j = 0; j < NJ; ++j) { f2r[j] = f2[j]; p2r[j] = p2[j]; }

    // ---------------- GEMM1: u[16x32] = x_tile[16x1536] @ W1[1536x32] ----------------
    v8f acc0 = {};   // N-tile 0 (cols 0..15)
    v8f acc1 = {};   // N-tile 1 (cols 16..31)

    for (int kb = 0; kb < DM / KB; ++kb) {
        const int k0 = kb * KB;

        // cooperative stage + transpose: W1[k0:k0+KB][0:32] -> lds_wt[n][k]
        // loads: 32 consecutive lanes read 32 consecutive n at fixed k (128B coalesced)
        // stores: banks 4*lane+k -> 2-way conflict only
#pragma unroll
        for (int it = 0; it < 16; ++it) {
            int q = it * 256 + tid;            // 4096 b32 total
            int n = q & 31, k = q >> 5;
            lds_wt[n * WTSTR + k] = w1[(size_t)(k0 + k) * N1 + n];
        }
        // per-wave stage: x[tok0:tok0+16][k0:k0+KB] -> lds_x (coalesced b128)
#pragma unroll
        for (int it = 0; it < 16; ++it) {
            int q   = it * 32 + lane;          // 512 float4 total
            int row = q >> 5, c4 = q & 31;
            v4f v = *(const v4f*)(x + (size_t)(tok0 + row) * DM + k0 + c4 * 4);
            *(v4f*)(xw + row * XSTR + c4 * 4) = v;
        }
        // prefetch next x k-block (global_prefetch_b8); 64 lines cover 16x128 f32
        if (kb + 1 < DM / KB) {
#pragma unroll
            for (int pf = 0; pf < 2; ++pf) {
                __builtin_prefetch(x + (size_t)(tok0 + l16) * DM + k0 + KB +
                                       (half * 2 + pf) * 32, 0, 0);
            }
        }
        __syncthreads();

#pragma unroll 4
        for (int ks = 0; ks < KB / 4; ++ks) {
            const int kk = ks * 4;
            // A 16x4 f32 frag: lane -> M=l16, VGPR pair = K {kk+2*half, kk+1+2*half}
            v2f a  = *(const v2f*)(xw + l16 * XSTR + kk + 2 * half);
            // B 4x16 f32 frags from transposed LDS: contiguous pair -> ds_load_b64
            v2f b0 = *(const v2f*)(lds_wt + l16 * WTSTR        + kk + 2 * half);
            v2f b1 = *(const v2f*)(lds_wt + (16 + l16) * WTSTR + kk + 2 * half);
            acc0 = __builtin_amdgcn_wmma_f32_16x16x4_f32(false, a, false, b0,
                                                         (short)0, acc0, false, false);
            acc1 = __builtin_amdgcn_wmma_f32_16x16x4_f32(false, a, false, b1,
                                                         (short)0, acc1, false, false);
        }
        __syncthreads();   // before next k-block overwrites lds_wt / lds_x
    }

    // ---------------- gelu-combine: g[t,i] = sum_j gelu(u*f2[j])*p2[j] ----------------
    // C/D layout: VGPR v -> row M = v + 8*half, col N = l16 (+16 for tile 1)
#pragma unroll
    for (int v = 0; v < 8; ++v) {
        float u0 = acc0[v], u1 = acc1[v];
        float g0 = 0.f, g1 = 0.f;
#pragma unroll
        for (int j = 0; j < NJ; ++j) {
            g0 += gelu_erf(u0 * f2r[j]) * p2r[j];
            g1 += gelu_erf(u1 * f2r[j]) * p2r[j];
        }
        const int row = v + 8 * half;
        gw[row * GSTR + l16]      = g0;
        gw[row * GSTR + 16 + l16] = g1;
    }

    // ---------------- GEMM2: out[16x1536] = G[16x32] @ P[32x1536] ----------------
    // whole A (16x32) as 8 k-step fragments; per-wave LDS is in-order -> no barrier
    v2f a2[8];
#pragma unroll
    for (int ks = 0; ks < 8; ++ks)
        a2[ks] = *(const v2f*)(gw + l16 * GSTR + ks * 4 + 2 * half);

    for (int nb = 0; nb < DM / NBLK; ++nb) {
        const int n0 = nb * NBLK;
        // cooperative stage + transpose: P[0:32][n0:n0+NBLK] -> lds_pt[n][k]
        // loads: 32 consecutive lanes read 32 consecutive n at fixed k (128B coalesced)
        // stores: banks 36*lane+k -> 2-way conflict only
#pragma unroll
        for (int it = 0; it < 16; ++it) {
            int q = it * 256 + tid;            // 4096 b32 total
            int n = q & 127, k = q >> 7;
            lds_pt[n * PTSTR + k] = p1[(size_t)k * DM + n0 + n];
        }
        __syncthreads();

#pragma unroll
        for (int nt = 0; nt < NBLK / 16; ++nt) {
            v8f c = {};
#pragma unroll
            for (int ks = 0; ks < 8; ++ks) {
                const int kk = ks * 4;
                // B frag: contiguous k pair in transposed LDS -> ds_load_b64
                v2f b = *(const v2f*)(lds_pt + (nt * 16 + l16) * PTSTR + kk + 2 * half);
                c = __builtin_amdgcn_wmma_f32_16x16x4_f32(false, a2[ks], false, b,
                                                          (short)0, c, false, false);
            }
            // coalesced fp32 stores: lanes 0-15 cover 64B runs per row
#pragma unroll
            for (int v = 0; v < 8; ++v) {
                out[(size_t)(tok0 + v + 8 * half) * DM + n0 + nt * 16 + l16] = c[v];
            }
        }
        __syncthreads();   // before next n-block overwrites lds_pt
    }
}

extern "C" void kernel_launch(void* const* d_in, const int* in_sizes, int n_in,
                              void* d_out, int out_size, void* d_ws, size_t ws_size,
                              hipStream_t stream) {
    const float* x  = (const float*)d_in[0];   // [16,4096,1536]
    const float* w1 = (const float*)d_in[1];   // c_fc_1  [1536,32]
    const float* f2 = (const float*)d_in[2];   // c_fc_2  [1,12]
    const float* p1 = (const float*)d_in[3];   // c_proj_1 [32,1536]
    const float* p2 = (const float*)d_in[4];   // c_proj_2 [12,1]
    float* out = (float*)d_out;

    const int tokens = in_sizes[0] / DM;               // 65536
    const int grid   = tokens / (16 * WAVES);          // 512 workgroups
    const size_t shmem = (size_t)(WAVES * 16 * XSTR + N1 * WTSTR +
                                  WAVES * 16 * GSTR + NBLK * PTSTR) * sizeof(float);

    krony_mlp_kernel<<<grid, 256, shmem, stream>>>(x, w1, f2, p1, p2, out);
}